// TRACE_86732569575520
// MI455X (gfx1250) — compile-verified
//
#include <hip/hip_runtime.h>
#include <hip/hip_bf16.h>

typedef float v2f __attribute__((ext_vector_type(2)));
typedef float v8f __attribute__((ext_vector_type(8)));

#define B_SZ 256
#define N_SZ 4096
#define C_SZ 256
#define R_SZ 32
#define K_SZ 128
#define LN2F 0.69314718055994530942f

// ---------------------------------------------------------------------------
// Kernel 1: per-row top-K (unordered) via byte-wise radix select on float keys
// ---------------------------------------------------------------------------
__device__ __forceinline__ unsigned f2key(float f) {
  unsigned u = __float_as_uint(f);
  return (u & 0x80000000u) ? ~u : (u | 0x80000000u);
}

__global__ void topk_kernel(const float* __restrict__ A,
                            float* __restrict__ vals,
                            int* __restrict__ idxo) {
  const int b = blockIdx.x;
  const int tid = threadIdx.x;
  const float* row = A + (size_t)b * N_SZ;

  __shared__ unsigned hist[256];
  __shared__ unsigned s_prefix;
  __shared__ int s_remaining;
  __shared__ unsigned s_cnt[2];

  if (tid == 0) { s_prefix = 0u; s_remaining = K_SZ; }
  __syncthreads();

  for (int byt = 3; byt >= 0; --byt) {
    hist[tid] = 0u;
    __syncthreads();
    unsigned prefix = s_prefix;
    unsigned hs = (unsigned)(byt + 1) * 8u;
    for (int i = tid; i < N_SZ; i += 256) {
      unsigned key = f2key(row[i]);
      bool match = (hs >= 32u) ? true : ((key >> hs) == (prefix >> hs));
      if (match) atomicAdd(&hist[(key >> (byt * 8)) & 0xFFu], 1u);
    }
    __syncthreads();
    if (tid == 0) {
      int rem = s_remaining;
      int cum = 0;
      for (int bin = 255; bin >= 0; --bin) {
        int cnt = (int)hist[bin];
        cum += cnt;
        if (cum >= rem) {
          s_prefix = s_prefix | ((unsigned)bin << (byt * 8));
          s_remaining = rem - (cum - cnt);
          break;
        }
      }
    }
    __syncthreads();
  }

  if (tid < 2) s_cnt[tid] = 0u;
  __syncthreads();
  const unsigned T = s_prefix;
  const int needEq = s_remaining;       // # of ==T elements to keep
  const int nGt = K_SZ - needEq;        // # of  >T elements (all kept)

  for (int i = tid; i < N_SZ; i += 256) {
    float f = row[i];
    unsigned key = f2key(f);
    if (key > T) {
      unsigned pos = atomicAdd(&s_cnt[0], 1u);
      vals[(size_t)b * K_SZ + pos] = f;
      idxo[(size_t)b * K_SZ + pos] = i;
    } else if (key == T) {
      unsigned e = atomicAdd(&s_cnt[1], 1u);
      if ((int)e < needEq) {
        int pos = nGt + (int)e;
        vals[(size_t)b * K_SZ + pos] = f;
        idxo[(size_t)b * K_SZ + pos] = i;
      }
    }
  }
}

// ---------------------------------------------------------------------------
// Kernel 2: PW = softplus(W) - ln2
// ---------------------------------------------------------------------------
__global__ void softplus_kernel(const float* __restrict__ W,
                                float* __restrict__ PW, int n) {
  int i = blockIdx.x * 256 + threadIdx.x;
  if (i < n) {
    float x = W[i];
    float sp = (x > 20.0f) ? x : log1pf(expf(x));
    PW[i] = sp - LN2F;
  }
}

// ---------------------------------------------------------------------------
// Kernel 3: fp32 WMMA NT-GEMM partials: Part[z][m,n] += X[m,:k]*Y[n,:k]
// Block: 256 thr / 8 waves. Each wave owns a 32x32 tile (2x2 wmma fragments,
// 4 independent accumulator chains) over its private K slice; 8 waves split K
// within a block, grid.z splits K across blocks (deterministic second-stage
// reduction, no float atomics). v_wmma_f32_16x16x4_f32 on the matrix pipe.
// ---------------------------------------------------------------------------
#define WMMA4(a, b, c)                                                        \
  __builtin_amdgcn_wmma_f32_16x16x4_f32(false, (a), false, (b), (short)0,     \
                                        (c), false, false)

__global__ void wmma_nt_gemm_f32(const float* __restrict__ X,
                                 const float* __restrict__ Y,
                                 float* __restrict__ Part,
                                 int ld, int Kdim, int Nn) {
  const int m0 = blockIdx.y * 32;
  const int n0 = blockIdx.x * 32;
  const int z = blockIdx.z;
  const int nz = gridDim.z;
  const int tid = threadIdx.x;
  const int wave = __builtin_amdgcn_readfirstlane(tid >> 5);  // scalar loop bounds
  const int lane = tid & 31;
  const int half = lane >> 4;  // 0: K+0..1, 1: K+2..3
  const int lm = lane & 15;

  const int kper = Kdim / (nz * 8);      // divisible by 8 for all our calls
  const int k0 = (z * 8 + wave) * kper;

  const float* xr0 = X + (size_t)(m0 + lm) * ld;
  const float* xr1 = X + (size_t)(m0 + 16 + lm) * ld;
  const float* yr0 = Y + (size_t)(n0 + lm) * ld;
  const float* yr1 = Y + (size_t)(n0 + 16 + lm) * ld;

  v8f acc00 = {}, acc01 = {}, acc10 = {}, acc11 = {};
  for (int kk = k0; kk < k0 + kper; kk += 8) {
    if ((kk & 31) == 0) {  // uniform condition; every 4th iteration
      // locality 3 -> near(WGP)-scope prefetch: pull into all cache levels
      __builtin_prefetch(xr0 + kk + 256, 0, 3);
      __builtin_prefetch(xr1 + kk + 256, 0, 3);
      __builtin_prefetch(yr0 + kk + 256, 0, 3);
      __builtin_prefetch(yr1 + kk + 256, 0, 3);
    }
    const int o0 = kk + half * 2;
    const int o1 = o0 + 4;
    v2f a00 = *(const v2f*)(xr0 + o0);
    v2f a10 = *(const v2f*)(xr1 + o0);
    v2f b00 = *(const v2f*)(yr0 + o0);
    v2f b10 = *(const v2f*)(yr1 + o0);
    v2f a01 = *(const v2f*)(xr0 + o1);
    v2f a11 = *(const v2f*)(xr1 + o1);
    v2f b01 = *(const v2f*)(yr0 + o1);
    v2f b11 = *(const v2f*)(yr1 + o1);
    acc00 = WMMA4(a00, b00, acc00);
    acc01 = WMMA4(a00, b10, acc01);
    acc10 = WMMA4(a10, b00, acc10);
    acc11 = WMMA4(a10, b10, acc11);
    acc00 = WMMA4(a01, b01, acc00);
    acc01 = WMMA4(a01, b11, acc01);
    acc10 = WMMA4(a11, b01, acc10);
    acc11 = WMMA4(a11, b11, acc11);
  }

  __shared__ float red[8][4][32][8];
#pragma unroll
  for (int v = 0; v < 8; ++v) {
    red[wave][0][lane][v] = acc00[v];
    red[wave][1][lane][v] = acc01[v];
    red[wave][2][lane][v] = acc10[v];
    red[wave][3][lane][v] = acc11[v];
  }
  __syncthreads();

  // 256 threads: each reduces 4 fragment cells over the 8 waves.
  const int L = tid & 31;
  const int v = tid >> 5;
  float* part = Part + (size_t)z * C_SZ * Nn;
#pragma unroll
  for (int f = 0; f < 4; ++f) {
    float s = 0.0f;
#pragma unroll
    for (int w = 0; w < 8; ++w) s += red[w][f][L][v];
    const int m = m0 + (f >> 1) * 16 + v + (L >> 4) * 8;  // C/D layout
    const int n = n0 + (f & 1) * 16 + (L & 15);
    part[(size_t)m * Nn + n] = s;
  }
}

// Deterministic K-slice reduction: Out[i] = sum_z Part[z][i]
__global__ void reduce_slices(const float* __restrict__ P,
                              float* __restrict__ Out, int n, int nz) {
  int i = blockIdx.x * 256 + threadIdx.x;
  if (i < n) {
    float s = 0.0f;
    for (int z = 0; z < nz; ++z) s += P[(size_t)z * n + i];
    Out[i] = s;
  }
}

// ---------------------------------------------------------------------------
// Kernel 4: Au[b,r,c] = sum_k vals[b,k] * U[c, idx[b,k], r]  (+ score_coact)
// ---------------------------------------------------------------------------
__global__ void au_kernel(const float* __restrict__ U,
                          const float* __restrict__ vals,
                          const int* __restrict__ idx,
                          float* __restrict__ Au,
                          float* __restrict__ score_coact) {
  const int b = blockIdx.x;
  const int c = threadIdx.x;
  __shared__ float sv[K_SZ];
  __shared__ int si[K_SZ];
  if (c < K_SZ) {
    sv[c] = vals[(size_t)b * K_SZ + c];
    si[c] = idx[(size_t)b * K_SZ + c];
  }
  __syncthreads();

  float4 acc[8];
#pragma unroll
  for (int j = 0; j < 8; ++j) acc[j] = make_float4(0.f, 0.f, 0.f, 0.f);

  const float* Uc = U + (size_t)c * N_SZ * R_SZ;
  for (int k = 0; k < K_SZ; ++k) {
    float v = sv[k];
    const float4* p = (const float4*)(Uc + (size_t)si[k] * R_SZ);
#pragma unroll
    for (int j = 0; j < 8; ++j) {
      float4 u4 = p[j];
      acc[j].x = fmaf(v, u4.x, acc[j].x);
      acc[j].y = fmaf(v, u4.y, acc[j].y);
      acc[j].z = fmaf(v, u4.z, acc[j].z);
      acc[j].w = fmaf(v, u4.w, acc[j].w);
    }
  }

  float ss = 0.0f;
  float* Aub = Au + (size_t)b * R_SZ * C_SZ;
#pragma unroll
  for (int j = 0; j < 8; ++j) {
    Aub[(j * 4 + 0) * C_SZ + c] = acc[j].x; ss = fmaf(acc[j].x, acc[j].x, ss);
    Aub[(j * 4 + 1) * C_SZ + c] = acc[j].y; ss = fmaf(acc[j].y, acc[j].y, ss);
    Aub[(j * 4 + 2) * C_SZ + c] = acc[j].z; ss = fmaf(acc[j].z, acc[j].z, ss);
    Aub[(j * 4 + 3) * C_SZ + c] = acc[j].w; ss = fmaf(acc[j].w, acc[j].w, ss);
  }
  score_coact[(size_t)b * C_SZ + c] = ss * (1.0f / ((float)K_SZ * R_SZ + 1e-9f));
}

// ---------------------------------------------------------------------------
// Kernel 5: per-b finalize: match -> softmax beta -> cross / beta^T G beta ->
// residual. One block per b, thread = c.
// ---------------------------------------------------------------------------
__global__ void finalize_kernel(const float* __restrict__ score_w,
                                const float* __restrict__ score_coact,
                                const float* __restrict__ span_t,
                                const float* __restrict__ proto_span,
                                const float* __restrict__ Au,
                                const float* __restrict__ G,
                                const float* __restrict__ vals,
                                float* __restrict__ out) {
  const float invKR = 1.0f / ((float)K_SZ * R_SZ + 1e-9f);
  const float invK = 1.0f / ((float)K_SZ + 1e-9f);
  const int b = blockIdx.x;
  const int c = threadIdx.x;

  __shared__ float sred[C_SZ];
  __shared__ float sbeta[C_SZ];
  __shared__ float s_a2[R_SZ];

  float span_norm = span_t[b] * (1.0f / 20.0f);
  span_norm = fminf(fmaxf(span_norm, 0.0f), 1.0f);
  float ps = 1.0f / (1.0f + expf(-proto_span[c]));
  // Row-mean subtraction of score_w omitted: softmax is shift-invariant per b.
  float match = score_w[(size_t)b * C_SZ + c] + score_coact[(size_t)b * C_SZ + c]
              - 0.3f * fabsf(span_norm - ps);

  // softmax over C
  sred[c] = match;
  __syncthreads();
  for (int s = C_SZ / 2; s > 0; s >>= 1) {
    if (c < s) sred[c] = fmaxf(sred[c], sred[c + s]);
    __syncthreads();
  }
  float mx = sred[0];
  __syncthreads();
  float e = expf(match - mx);
  sred[c] = e;
  __syncthreads();
  for (int s = C_SZ / 2; s > 0; s >>= 1) {
    if (c < s) sred[c] += sred[c + s];
    __syncthreads();
  }
  float denom = sred[0];
  __syncthreads();
  float beta = e / denom;
  sbeta[c] = beta;
  __syncthreads();

  // Au2[r] = sum_c beta[c] * Au[b,r,c]; 8 threads per r, 32 c's each
  const int r = c >> 3;
  const int part = c & 7;
  const float* Aub = Au + (size_t)b * R_SZ * C_SZ + (size_t)r * C_SZ;
  float p = 0.0f;
  for (int j = part * 32; j < part * 32 + 32; ++j)
    p = fmaf(sbeta[j], Aub[j], p);
  sred[c] = p;
  __syncthreads();
  if (part == 0) {
    float a2 = 0.0f;
#pragma unroll
    for (int q = 0; q < 8; ++q) a2 += sred[c + q];
    s_a2[r] = a2;
  }
  __syncthreads();

  // (G beta)[c] using G symmetry for coalesced column reads
  float gb = 0.0f;
  for (int cp = 0; cp < C_SZ; ++cp)
    gb = fmaf(G[(size_t)cp * C_SZ + c], sbeta[cp], gb);

  // residual is linear in the three sums -> single tree reduction
  float contrib = beta * gb * invKR;                        // norm_Mrecon part
  if (c < K_SZ) {
    float v = vals[(size_t)b * K_SZ + c];
    contrib = fmaf(v * v, invK, contrib);                   // norm_Mhat_n part
  }
  if (c < R_SZ) {
    float a2 = s_a2[c];
    contrib -= 2.0f * invKR * a2 * a2;                      // -2*cross part
  }
  sred[c] = contrib;
  __syncthreads();
  for (int s = C_SZ / 2; s > 0; s >>= 1) {
    if (c < s) sred[c] += sred[c + s];
    __syncthreads();
  }
  if (c == 0) {
    float resid = sred[0];
    out[b] = fminf(fmaxf(resid, 0.0f), 10000.0f);
  }
}

// ---------------------------------------------------------------------------
extern "C" void kernel_launch(void* const* d_in, const int* in_sizes, int n_in,
                              void* d_out, int out_size, void* d_ws, size_t ws_size,
                              hipStream_t stream) {
  const float* A      = (const float*)d_in[0];  // [B,N]
  const float* span_t = (const float*)d_in[1];  // [B]
  const float* W      = (const float*)d_in[2];  // [C,N]
  const float* U      = (const float*)d_in[3];  // [C,N,R]
  const float* pspan  = (const float*)d_in[4];  // [C]
  float* out = (float*)d_out;                   // [B]

  char* ws = (char*)d_ws;
  size_t off = 0;
  float* vals = (float*)(ws + off);   off += (size_t)B_SZ * K_SZ * sizeof(float);
  int* idx    = (int*)(ws + off);     off += (size_t)B_SZ * K_SZ * sizeof(int);
  float* PW   = (float*)(ws + off);   off += (size_t)C_SZ * N_SZ * sizeof(float);
  float* sw   = (float*)(ws + off);   off += (size_t)B_SZ * C_SZ * sizeof(float);
  float* sc   = (float*)(ws + off);   off += (size_t)B_SZ * C_SZ * sizeof(float);
  float* Au   = (float*)(ws + off);   off += (size_t)B_SZ * R_SZ * C_SZ * sizeof(float);
  float* G    = (float*)(ws + off);   off += (size_t)C_SZ * C_SZ * sizeof(float);
  float* Ps   = (float*)(ws + off);   off += (size_t)2 * C_SZ * C_SZ * sizeof(float);
  float* Pg   = (float*)(ws + off);   off += (size_t)8 * C_SZ * C_SZ * sizeof(float);
  (void)ws_size; (void)in_sizes; (void)n_in; (void)out_size;

  const int nCC = C_SZ * C_SZ;  // 65536

  // 1) top-K per row
  topk_kernel<<<B_SZ, 256, 0, stream>>>(A, vals, idx);

  // 2) PW = softplus(W) - ln2
  const int nW = C_SZ * N_SZ;
  softplus_kernel<<<(nW + 255) / 256, 256, 0, stream>>>(W, PW, nW);

  // 3) score_w = A * PW^T  [256x256, inner 4096], K split 2x8 ways
  wmma_nt_gemm_f32<<<dim3(C_SZ / 32, B_SZ / 32, 2), 256, 0, stream>>>(
      A, PW, Ps, N_SZ, N_SZ, C_SZ);
  reduce_slices<<<(nCC + 255) / 256, 256, 0, stream>>>(Ps, sw, nCC, 2);

  // 4) Gram G = Uflat*Uflat^T  [256x256, inner 131072], K split 8x8 ways
  wmma_nt_gemm_f32<<<dim3(C_SZ / 32, C_SZ / 32, 8), 256, 0, stream>>>(
      U, U, Pg, N_SZ * R_SZ, N_SZ * R_SZ, C_SZ);
  reduce_slices<<<(nCC + 255) / 256, 256, 0, stream>>>(Pg, G, nCC, 8);

  // 5) Au + score_coact (gathers hit L2-resident U)
  au_kernel<<<B_SZ, 256, 0, stream>>>(U, vals, idx, Au, sc);

  // 6) finalize -> residual[b]
  finalize_kernel<<<B_SZ, 256, 0, stream>>>(sw, sc, span_t, pspan, Au, G, vals, out);
}